// GNNHetero_60885456389013
// MI455X (gfx1250) — compile-verified
//
#include <hip/hip_runtime.h>
#include <cstddef>

typedef __attribute__((ext_vector_type(2))) float v2f;
typedef __attribute__((ext_vector_type(8))) float v8f;

#define GNN_N 100000
#define GNN_E 600000
#define GNN_H 128
#define GNN_G 64
#define TKC 64          // K-chunk staged in LDS (64x128 f32 = 32KB)

// ---------------------------------------------------------------------------
// utility kernels
// ---------------------------------------------------------------------------
__global__ void fill_u32_kernel(unsigned int* __restrict__ p, unsigned int v, long long n) {
    long long i = (long long)blockIdx.x * blockDim.x + threadIdx.x;
    long long stride = (long long)gridDim.x * blockDim.x;
    for (; i < n; i += stride) p[i] = v;
}

__global__ void add2_kernel(const float* __restrict__ a, const float* __restrict__ b,
                            float* __restrict__ o, int n) {
    int i = blockIdx.x * blockDim.x + threadIdx.x;
    if (i < n) o[i] = a[i] + b[i];
}

// ---------------------------------------------------------------------------
// scatter-add: agg[dst] += x[src]  (D = 64 or 128, float4 per thread)
// ---------------------------------------------------------------------------
__global__ void scatter_add_kernel(const float* __restrict__ x, const int* __restrict__ ei,
                                   float* __restrict__ agg, int E, int D) {
    int idx = blockIdx.x * blockDim.x + threadIdx.x;
    int vecs = D >> 2;
    int e = idx / vecs;
    int c = idx - e * vecs;
    if (e >= E) return;
    int src = ei[e];
    int dst = ei[E + e];
    const float4 v = ((const float4*)(x + (size_t)src * D))[c];
    float* d = agg + (size_t)dst * D + c * 4;
    atomicAdd(d + 0, v.x);
    atomicAdd(d + 1, v.y);
    atomicAdd(d + 2, v.z);
    atomicAdd(d + 3, v.w);
}

// ---------------------------------------------------------------------------
// fused multi-product GEMM:  out[M x 128] = sum_p X_p[M x K] @ W_p[K x 128]
//                                           + bias, optional ReLU
// one wave -> 16 rows x 128 cols via 8x V_WMMA_F32_16X16X4_F32 per k-step
// ---------------------------------------------------------------------------
__global__ __launch_bounds__(256) void gemm_fused_kernel(
    float* __restrict__ out,
    const float* __restrict__ X0, const float* __restrict__ W0,
    const float* __restrict__ X1, const float* __restrict__ W1,
    const float* __restrict__ X2, const float* __restrict__ W2,
    const float* __restrict__ bias, int M, int K, int relu)
{
    __shared__ float sW[TKC * GNN_H];     // 32 KB

    const int lane = threadIdx.x & 31;
    const int wave = threadIdx.x >> 5;
    const int row0 = (blockIdx.x * 8 + wave) * 16;
    const bool active = row0 < M;          // wave-uniform -> EXEC all-1s inside
    const int lo = lane & 15;
    const int hi = lane >> 4;              // 0 or 1
    const int koff = hi * 2;               // A/B K sub-offset per ISA layout

    v8f acc[8] = {};

    const float* Xs[3] = {X0, X1, X2};
    const float* Ws[3] = {W0, W1, W2};

    #pragma unroll
    for (int p = 0; p < 3; ++p) {
        const float* X = Xs[p];
        const float* W = Ws[p];
        if (X == nullptr) continue;        // uniform across block
        for (int kc = 0; kc < K; kc += TKC) {
            __syncthreads();               // protect previous chunk's LDS reads
            for (int i = threadIdx.x; i < (TKC * GNN_H) / 4; i += 256)
                ((float4*)sW)[i] = ((const float4*)(W + (size_t)kc * GNN_H))[i];
            __syncthreads();
            if (active) {
                const float* Arow = X + (size_t)(row0 + lo) * K + kc + koff;
                #pragma unroll
                for (int kk = 0; kk < TKC; kk += 4) {
                    // A 16x4 fragment: lanes0-15 K=kk+0,1 ; lanes16-31 K=kk+2,3
                    const v2f a = *reinterpret_cast<const v2f*>(Arow + kk);
                    #pragma unroll
                    for (int t = 0; t < 8; ++t) {
                        v2f b;
                        b.x = sW[(kk + koff)     * GNN_H + t * 16 + lo];
                        b.y = sW[(kk + koff + 1) * GNN_H + t * 16 + lo];
                        acc[t] = __builtin_amdgcn_wmma_f32_16x16x4_f32(
                            false, a, false, b, (short)0, acc[t], false, false);
                    }
                }
            }
        }
    }

    if (active) {
        #pragma unroll
        for (int t = 0; t < 8; ++t) {
            const int col = t * 16 + lo;
            const float bv = bias ? bias[col] : 0.0f;
            #pragma unroll
            for (int r = 0; r < 8; ++r) {
                float v = acc[t][r] + bv;
                if (relu) v = fmaxf(v, 0.0f);
                // C/D layout: VGPR r -> row r (lanes0-15) / r+8 (lanes16-31)
                out[(size_t)(row0 + hi * 8 + r) * GNN_H + col] = v;
            }
        }
    }
}

// ---------------------------------------------------------------------------
// segment-max with sorted batch ids (flush one atomic per batch change)
// ---------------------------------------------------------------------------
__device__ inline void atomicMaxFloat(float* addr, float v) {
    if (v >= 0.0f) atomicMax((int*)addr, __float_as_int(v));
    else           atomicMin((unsigned int*)addr, __float_as_uint(v));
}

__global__ void segmax_kernel(const float* __restrict__ h, const int* __restrict__ batch,
                              float* __restrict__ gm, int n) {
    int tid = blockIdx.x * blockDim.x + threadIdx.x;
    int col = tid & (GNN_H - 1);
    int n0 = (tid >> 7) * 16;
    if (n0 >= n) return;
    int nend = n0 + 16 < n ? n0 + 16 : n;
    float m = -3.402823466e38f;
    int curb = -1;
    for (int nn = n0; nn < nend; ++nn) {
        int b = batch[nn];
        if (b != curb) {
            if (curb >= 0) atomicMaxFloat(&gm[curb * GNN_H + col], m);
            curb = b;
            m = -3.402823466e38f;
        }
        m = fmaxf(m, h[(size_t)nn * GNN_H + col]);
    }
    if (curb >= 0) atomicMaxFloat(&gm[curb * GNN_H + col], m);
}

// ---------------------------------------------------------------------------
// head: per-graph MLPs + final combine; one thread per graph
// ---------------------------------------------------------------------------
__global__ void head_kernel(const float* __restrict__ gm_pe, const float* __restrict__ gm_r,
                            const float* __restrict__ W1, const float* __restrict__ b1,
                            const float* __restrict__ W2, const float* __restrict__ b2,
                            const float* __restrict__ outW, const float* __restrict__ outb,
                            float* __restrict__ out) {
    int g = threadIdx.x;
    if (g >= GNN_G) return;
    float o[2];
    #pragma unroll
    for (int t = 0; t < 2; ++t) {
        const float* gm = (t == 0 ? gm_pe : gm_r) + g * GNN_H;
        float hb[5];
        #pragma unroll
        for (int j = 0; j < 5; ++j) {
            float s = b1[t * 5 + j];
            for (int k = 0; k < GNN_H; ++k) s += gm[k] * W1[t * GNN_H * 5 + k * 5 + j];
            hb[j] = fmaxf(s, 0.0f);
        }
        float s2 = b2[t];
        #pragma unroll
        for (int j = 0; j < 5; ++j) s2 += hb[j] * W2[t * 5 + j];
        o[t] = s2;
    }
    out[g] = o[0] * outW[0] + o[1] * outW[1] + outb[0];
}

// ---------------------------------------------------------------------------
// host-side orchestration
// ---------------------------------------------------------------------------
static inline void zero_buf(float* p, long long n, hipStream_t s) {
    fill_u32_kernel<<<2048, 256, 0, s>>>((unsigned int*)p, 0u, n);
}

static inline void launch_gemm(hipStream_t s, float* out,
                               const float* X0, const float* W0,
                               const float* X1, const float* W1,
                               const float* X2, const float* W2,
                               const float* bias, int M, int K, int relu) {
    int rowTiles = (M + 15) / 16;
    int blocks = (rowTiles + 7) / 8;
    gemm_fused_kernel<<<blocks, 256, 0, s>>>(out, X0, W0, X1, W1, X2, W2, bias, M, K, relu);
}

static inline void launch_scatter(hipStream_t s, const float* x, const int* ei,
                                  float* agg, int D) {
    long long threads = (long long)GNN_E * (D / 4);
    int blocks = (int)((threads + 255) / 256);
    scatter_add_kernel<<<blocks, 256, 0, s>>>(x, ei, agg, GNN_E, D);
}

extern "C" void kernel_launch(void* const* d_in, const int* in_sizes, int n_in,
                              void* d_out, int out_size, void* d_ws, size_t ws_size,
                              hipStream_t stream) {
    (void)in_sizes; (void)n_in; (void)out_size; (void)ws_size;

    const float* x_pe   = (const float*)d_in[0];
    const float* x_r    = (const float*)d_in[1];
    const int*   ei_per = (const int*)d_in[2];
    const int*   ei_rpe = (const int*)d_in[3];
    const int*   ei_rr  = (const int*)d_in[4];
    const int*   bpe    = (const int*)d_in[5];
    const int*   br     = (const int*)d_in[6];
    const float* Wrel1  = (const float*)d_in[7];    // [3,64,128]
    const float* brel1  = (const float*)d_in[8];    // [3,128]
    const float* Wroot1 = (const float*)d_in[9];    // [3,64,128]
    const float* Wrel23 = (const float*)d_in[10];   // [2,3,128,128]
    const float* brel23 = (const float*)d_in[11];   // [2,3,128]
    const float* Wroot23= (const float*)d_in[12];   // [2,3,128,128]
    const float* mW1    = (const float*)d_in[13];
    const float* mb1    = (const float*)d_in[14];
    const float* mW2    = (const float*)d_in[15];
    const float* mb2    = (const float*)d_in[16];
    const float* outW   = (const float*)d_in[17];
    const float* outb   = (const float*)d_in[18];
    float* out = (float*)d_out;

    const int N = GNN_N;
    const size_t NB = (size_t)N * GNN_H;
    float* ws = (float*)d_ws;
    float* B0 = ws;              // rotating node buffers, each N x 128
    float* B1 = B0 + NB;
    float* B2 = B1 + NB;
    float* B3 = B2 + NB;
    float* B4 = B3 + NB;
    float* Wsum  = B4 + NB;      // 128*128
    float* bsum  = Wsum + 128 * 128;
    float* gm_pe = bsum + 128;   // 64*128
    float* gm_r  = gm_pe + GNN_G * GNN_H;

    // ------------------------- Layer 1 (K = 64) ---------------------------
    zero_buf(B0, (long long)N * 64, stream);
    zero_buf(B1, (long long)N * 64, stream);
    zero_buf(B2, (long long)N * 64, stream);
    launch_scatter(stream, x_r,  ei_rpe, B0, 64);   // agg_pe_from_r
    launch_scatter(stream, x_pe, ei_per, B1, 64);   // agg_r_from_pe
    launch_scatter(stream, x_r,  ei_rr,  B2, 64);   // agg_r_from_r
    add2_kernel<<<(64 * 128 + 255) / 256, 256, 0, stream>>>(Wroot1 + 0 * 64 * 128,
                                                            Wroot1 + 2 * 64 * 128, Wsum, 64 * 128);
    add2_kernel<<<1, 128, 0, stream>>>(brel1 + 0 * 128, brel1 + 2 * 128, bsum, 128);
    launch_gemm(stream, B3, B0, Wrel1 + 1 * 64 * 128, x_pe, Wroot1 + 1 * 64 * 128,
                nullptr, nullptr, brel1 + 1 * 128, N, 64, 1);                     // h_pe
    launch_gemm(stream, B4, B1, Wrel1 + 0 * 64 * 128, B2, Wrel1 + 2 * 64 * 128,
                x_r, Wsum, bsum, N, 64, 1);                                       // h_r

    // ------------------------- Layer 2 (K = 128) --------------------------
    const float* Wr2 = Wrel23;                  // [3,128,128]
    const float* Wo2 = Wroot23;
    const float* bb2 = brel23;
    zero_buf(B0, (long long)N * 128, stream);
    zero_buf(B1, (long long)N * 128, stream);
    zero_buf(B2, (long long)N * 128, stream);
    launch_scatter(stream, B4, ei_rpe, B0, 128);
    launch_scatter(stream, B3, ei_per, B1, 128);
    launch_scatter(stream, B4, ei_rr,  B2, 128);
    add2_kernel<<<(128 * 128 + 255) / 256, 256, 0, stream>>>(Wo2 + 0 * 128 * 128,
                                                             Wo2 + 2 * 128 * 128, Wsum, 128 * 128);
    add2_kernel<<<1, 128, 0, stream>>>(bb2 + 0 * 128, bb2 + 2 * 128, bsum, 128);
    // out overwrites its own A-input (safe: each wave writes only rows it consumed)
    launch_gemm(stream, B0, B0, Wr2 + 1 * 128 * 128, B3, Wo2 + 1 * 128 * 128,
                nullptr, nullptr, bb2 + 1 * 128, N, 128, 1);                      // h_pe -> B0
    launch_gemm(stream, B1, B1, Wr2 + 0 * 128 * 128, B2, Wr2 + 2 * 128 * 128,
                B4, Wsum, bsum, N, 128, 1);                                       // h_r -> B1

    // ------------------------- Layer 3 (K = 128, no ReLU) -----------------
    const float* Wr3 = Wrel23  + 3 * 128 * 128;
    const float* Wo3 = Wroot23 + 3 * 128 * 128;
    const float* bb3 = brel23  + 3 * 128;
    zero_buf(B2, (long long)N * 128, stream);
    zero_buf(B3, (long long)N * 128, stream);
    zero_buf(B4, (long long)N * 128, stream);
    launch_scatter(stream, B1, ei_rpe, B2, 128);    // agg_pe_from_r
    launch_scatter(stream, B0, ei_per, B3, 128);    // agg_r_from_pe
    launch_scatter(stream, B1, ei_rr,  B4, 128);    // agg_r_from_r
    add2_kernel<<<(128 * 128 + 255) / 256, 256, 0, stream>>>(Wo3 + 0 * 128 * 128,
                                                             Wo3 + 2 * 128 * 128, Wsum, 128 * 128);
    add2_kernel<<<1, 128, 0, stream>>>(bb3 + 0 * 128, bb3 + 2 * 128, bsum, 128);
    launch_gemm(stream, B2, B2, Wr3 + 1 * 128 * 128, B0, Wo3 + 1 * 128 * 128,
                nullptr, nullptr, bb3 + 1 * 128, N, 128, 0);                      // h_pe -> B2
    launch_gemm(stream, B3, B3, Wr3 + 0 * 128 * 128, B4, Wr3 + 2 * 128 * 128,
                B1, Wsum, bsum, N, 128, 0);                                       // h_r -> B3

    // ------------------------- segment-max + head -------------------------
    fill_u32_kernel<<<64, 256, 0, stream>>>((unsigned int*)gm_pe, 0xFF7FFFFFu,
                                            (long long)2 * GNN_G * GNN_H);        // -FLT_MAX
    {
        long long threads = (long long)GNN_H * ((N + 15) / 16);
        int blocks = (int)((threads + 255) / 256);
        segmax_kernel<<<blocks, 256, 0, stream>>>(B2, bpe, gm_pe, N);
        segmax_kernel<<<blocks, 256, 0, stream>>>(B3, br,  gm_r,  N);
    }
    head_kernel<<<1, 64, 0, stream>>>(gm_pe, gm_r, mW1, mb1, mW2, mb2, outW, outb, out);
}